// Expert_10153302688477
// MI455X (gfx1250) — compile-verified
//
#include <hip/hip_runtime.h>
#include <math.h>

typedef float v2f __attribute__((ext_vector_type(2)));
typedef float v8f __attribute__((ext_vector_type(8)));

#define E_EXPERTS 8
#define TOPK 2
#define DDIM 1024
#define HDIM 2048

#define BM 64          // token rows per workgroup tile (4 M sub-tiles per wave)
#define BN 256         // output cols per workgroup tile (8 waves x 32 cols)
#define KC 64          // K chunk staged in LDS
#define LDSTR 68       // padded LDS row stride (floats); 68*4 = 272 = 16*17 -> float4-aligned rows

// ---------------------------------------------------------------------------
// K0: zero the per-expert counters
// ---------------------------------------------------------------------------
__global__ void zero_counts_kernel(int* counts) {
    if (threadIdx.x < E_EXPERTS) counts[threadIdx.x] = 0;
}

// ---------------------------------------------------------------------------
// K1: gating. One wave per token: logits = x @ gate_w^T, top-2 (first index
// wins ties, matching jax top_k), 2-way softmax, append to expert lists.
// entry = token | (k << 16), k = rank.
// ---------------------------------------------------------------------------
__global__ __launch_bounds__(128)
void gate_kernel(const float* __restrict__ x, const float* __restrict__ gw,
                 int* __restrict__ counts, int* __restrict__ entries,
                 float* __restrict__ wlist, int T) {
    const int lane = threadIdx.x & 31;
    const int t = blockIdx.x * (blockDim.x >> 5) + (threadIdx.x >> 5);
    if (t >= T) return;

    float acc[E_EXPERTS];
#pragma unroll
    for (int e = 0; e < E_EXPERTS; ++e) acc[e] = 0.0f;

    for (int i = 0; i < DDIM / 32; ++i) {
        const int d = lane + 32 * i;
        const float xv = x[(size_t)t * DDIM + d];
#pragma unroll
        for (int e = 0; e < E_EXPERTS; ++e)
            acc[e] += xv * gw[e * DDIM + d];
    }
#pragma unroll
    for (int e = 0; e < E_EXPERTS; ++e) {
#pragma unroll
        for (int off = 16; off > 0; off >>= 1)
            acc[e] += __shfl_xor(acc[e], off, 32);
    }
    if (lane == 0) {
        int e0 = 0; float l0 = acc[0];
        for (int e = 1; e < E_EXPERTS; ++e)
            if (acc[e] > l0) { l0 = acc[e]; e0 = e; }
        int e1 = -1; float l1 = -3.0e38f;
        for (int e = 0; e < E_EXPERTS; ++e)
            if (e != e0 && acc[e] > l1) { l1 = acc[e]; e1 = e; }
        const float p1 = __expf(l1 - l0);
        const float s  = 1.0f + p1;
        const float w0 = 1.0f / s;
        const float w1 = p1 / s;
        int s0 = atomicAdd(&counts[e0], 1);
        entries[e0 * T + s0] = t;                 // k = 0
        wlist  [e0 * T + s0] = w0;
        int s1 = atomicAdd(&counts[e1], 1);
        entries[e1 * T + s1] = t | (1 << 16);     // k = 1
        wlist  [e1 * T + s1] = w1;
    }
}

// ---------------------------------------------------------------------------
// K2: exclusive prefix sum of 8 counts -> packed h-buffer row offsets
// ---------------------------------------------------------------------------
__global__ void scan_kernel(const int* __restrict__ counts, int* __restrict__ offsets) {
    if (threadIdx.x == 0 && blockIdx.x == 0) {
        int s = 0;
        for (int e = 0; e < E_EXPERTS; ++e) { offsets[e] = s; s += counts[e]; }
    }
}

// ---------------------------------------------------------------------------
// K3: h = (x@W0 + b0) * silu(x@W1 + b1), gathered token tiles.
// 8 waves; each wave owns a 64(M) x 32(N) slice of the 64x256 block tile:
// 4 M sub-tiles x 2 N sub-tiles x 2 matrices = 16 f32 WMMA accumulators.
// Each B fragment feeds four WMMAs -> 16 WMMA per 8 B loads per k-step.
// ---------------------------------------------------------------------------
__global__ __launch_bounds__(256)
void ffn1_kernel(const float* __restrict__ x,
                 const float* __restrict__ W0, const float* __restrict__ b0,
                 const float* __restrict__ W1, const float* __restrict__ b1,
                 const int* __restrict__ counts, const int* __restrict__ offsets,
                 const int* __restrict__ entries,
                 float* __restrict__ hbuf, int T) {
    const int e   = blockIdx.z;
    const int cnt = counts[e];
    const int mbase = blockIdx.y * BM;
    if (mbase >= cnt) return;
    const int hbase = blockIdx.x * BN;

    __shared__ float xs[BM][LDSTR];
    __shared__ int stok[BM];

    const int tid  = threadIdx.x;
    const int lane = tid & 31;
    const int wn   = tid >> 5;        // 0..7 : 32-col slice of BN
    const int half = lane >> 4;       // 0/1 : K sub-pair per A/B layout
    const int l16  = lane & 15;

    if (tid < BM) {
        int slot = mbase + tid;
        int s2 = slot < cnt ? slot : cnt - 1;
        stok[tid] = entries[e * T + s2] & 0xFFFF;
    }
    __syncthreads();

    const float* W0e = W0 + (size_t)e * DDIM * HDIM;
    const float* W1e = W1 + (size_t)e * DDIM * HDIM;
    const int colA = hbase + wn * 32 + l16;   // N sub-tile 0 column
    const int colB = colA + 16;               // N sub-tile 1 column
    const int oe = offsets[e];

    v8f acc0[4][2] = {};   // W0: [Mtile][Ntile]
    v8f acc1[4][2] = {};   // W1: [Mtile][Ntile]

    for (int kc = 0; kc < DDIM; kc += KC) {
        __syncthreads();
        // stage x tile (float4 -> ds_store_b128; rows are 16B aligned)
        for (int idx = tid; idx < BM * (KC / 4); idx += 256) {
            const int r  = idx >> 4;            // / (KC/4)
            const int c4 = (idx & 15) << 2;
            const float4 v = *(const float4*)&x[(size_t)stok[r] * DDIM + kc + c4];
            *(float4*)&xs[r][c4] = v;
        }
        __syncthreads();
        // prefetch next weight chunk toward near caches
        if (kc + KC < DDIM) {
            const size_t nk = (size_t)(kc + KC + 2 * half) * HDIM;
            __builtin_prefetch(W0e + nk + colA, 0, 1);
            __builtin_prefetch(W1e + nk + colA, 0, 1);
        }
#pragma unroll 2
        for (int kk = 0; kk < KC; kk += 4) {
            v2f am[4];
#pragma unroll
            for (int mt = 0; mt < 4; ++mt) {
                am[mt].x = xs[mt * 16 + l16][kk + 2 * half];
                am[mt].y = xs[mt * 16 + l16][kk + 2 * half + 1];
            }
            const int kg = kc + kk + 2 * half;
            const float* p0 = W0e + (size_t)kg * HDIM;
            const float* p1 = W1e + (size_t)kg * HDIM;
            v2f b;
            b.x = p0[colA]; b.y = p0[HDIM + colA];
#pragma unroll
            for (int mt = 0; mt < 4; ++mt)
                acc0[mt][0] = __builtin_amdgcn_wmma_f32_16x16x4_f32(false, am[mt], false, b, (short)0, acc0[mt][0], false, false);
            b.x = p0[colB]; b.y = p0[HDIM + colB];
#pragma unroll
            for (int mt = 0; mt < 4; ++mt)
                acc0[mt][1] = __builtin_amdgcn_wmma_f32_16x16x4_f32(false, am[mt], false, b, (short)0, acc0[mt][1], false, false);
            b.x = p1[colA]; b.y = p1[HDIM + colA];
#pragma unroll
            for (int mt = 0; mt < 4; ++mt)
                acc1[mt][0] = __builtin_amdgcn_wmma_f32_16x16x4_f32(false, am[mt], false, b, (short)0, acc1[mt][0], false, false);
            b.x = p1[colB]; b.y = p1[HDIM + colB];
#pragma unroll
            for (int mt = 0; mt < 4; ++mt)
                acc1[mt][1] = __builtin_amdgcn_wmma_f32_16x16x4_f32(false, am[mt], false, b, (short)0, acc1[mt][1], false, false);
        }
    }

    // epilogue: bias + SwiGLU, scatter to packed h rows
    const float b0a = b0[e * HDIM + colA], b0b = b0[e * HDIM + colB];
    const float b1a = b1[e * HDIM + colA], b1b = b1[e * HDIM + colB];
#pragma unroll
    for (int mt = 0; mt < 4; ++mt) {
#pragma unroll
        for (int j = 0; j < 8; ++j) {
            const int slot = mbase + mt * 16 + j + 8 * half;
            if (slot < cnt) {
                const size_t row = (size_t)(oe + slot) * HDIM;
                const float vA = acc1[mt][0][j] + b1a;
                const float vB = acc1[mt][1][j] + b1b;
                hbuf[row + colA] = (acc0[mt][0][j] + b0a) * (vA / (1.0f + __expf(-vA)));
                hbuf[row + colB] = (acc0[mt][1][j] + b0b) * (vB / (1.0f + __expf(-vB)));
            }
        }
    }
}

// ---------------------------------------------------------------------------
// K4: y = h @ W2 + b2, scaled by gate weight, scattered to pair_out[k][t][:].
// Same 64x32-per-wave tiling; 8 accumulators per wave.
// ---------------------------------------------------------------------------
__global__ __launch_bounds__(256)
void ffn2_kernel(const float* __restrict__ hbuf,
                 const float* __restrict__ W2, const float* __restrict__ b2,
                 const int* __restrict__ counts, const int* __restrict__ offsets,
                 const int* __restrict__ entries, const float* __restrict__ wlist,
                 float* __restrict__ pair_out, int T) {
    const int e   = blockIdx.z;
    const int cnt = counts[e];
    const int mbase = blockIdx.y * BM;
    if (mbase >= cnt) return;
    const int dbase = blockIdx.x * BN;

    __shared__ float as[BM][LDSTR];
    __shared__ int   sEnt[BM];
    __shared__ float sW[BM];

    const int tid  = threadIdx.x;
    const int lane = tid & 31;
    const int wn   = tid >> 5;
    const int half = lane >> 4;
    const int l16  = lane & 15;
    const int oe   = offsets[e];

    if (tid < BM) {
        int slot = mbase + tid;
        int s2 = slot < cnt ? slot : cnt - 1;
        sEnt[tid] = entries[e * T + s2];
        sW[tid]   = wlist  [e * T + s2];
    }
    __syncthreads();

    const float* W2e = W2 + (size_t)e * HDIM * DDIM;
    const int colA = dbase + wn * 32 + l16;
    const int colB = colA + 16;

    v8f acc[4][2] = {};   // [Mtile][Ntile]

    for (int kc = 0; kc < HDIM; kc += KC) {
        __syncthreads();
        for (int idx = tid; idx < BM * (KC / 4); idx += 256) {
            const int r  = idx >> 4;
            const int c4 = (idx & 15) << 2;
            const int slot = mbase + r;
            const int s2 = slot < cnt ? slot : cnt - 1;
            const float4 v = *(const float4*)&hbuf[(size_t)(oe + s2) * HDIM + kc + c4];
            *(float4*)&as[r][c4] = v;
        }
        __syncthreads();
        if (kc + KC < HDIM) {
            __builtin_prefetch(W2e + (size_t)(kc + KC + 2 * half) * DDIM + colA, 0, 1);
        }
#pragma unroll 2
        for (int kk = 0; kk < KC; kk += 4) {
            v2f am[4];
#pragma unroll
            for (int mt = 0; mt < 4; ++mt) {
                am[mt].x = as[mt * 16 + l16][kk + 2 * half];
                am[mt].y = as[mt * 16 + l16][kk + 2 * half + 1];
            }
            const int kg = kc + kk + 2 * half;
            const float* p = W2e + (size_t)kg * DDIM;
            v2f b;
            b.x = p[colA]; b.y = p[DDIM + colA];
#pragma unroll
            for (int mt = 0; mt < 4; ++mt)
                acc[mt][0] = __builtin_amdgcn_wmma_f32_16x16x4_f32(false, am[mt], false, b, (short)0, acc[mt][0], false, false);
            b.x = p[colB]; b.y = p[DDIM + colB];
#pragma unroll
            for (int mt = 0; mt < 4; ++mt)
                acc[mt][1] = __builtin_amdgcn_wmma_f32_16x16x4_f32(false, am[mt], false, b, (short)0, acc[mt][1], false, false);
        }
    }

    const float b2a = b2[e * DDIM + colA];
    const float b2b = b2[e * DDIM + colB];
#pragma unroll
    for (int mt = 0; mt < 4; ++mt) {
#pragma unroll
        for (int j = 0; j < 8; ++j) {
            const int slot = mbase + mt * 16 + j + 8 * half;
            if (slot < cnt) {
                const int   ent = sEnt[slot - mbase];
                const float wgt = sW[slot - mbase];
                const int t = ent & 0xFFFF;
                const int k = (ent >> 16) & 1;
                const size_t ro = ((size_t)k * T + t) * DDIM;
                pair_out[ro + colA] = wgt * (acc[mt][0][j] + b2a);
                pair_out[ro + colB] = wgt * (acc[mt][1][j] + b2b);
            }
        }
    }
}

// ---------------------------------------------------------------------------
// K5: out = pair_out[0] + pair_out[1]  (float4)
// ---------------------------------------------------------------------------
__global__ void combine_kernel(const float* __restrict__ pair_out,
                               float* __restrict__ out, int n4) {
    const float4* p = (const float4*)pair_out;
    float4* o = (float4*)out;
    for (int i = blockIdx.x * blockDim.x + threadIdx.x; i < n4;
         i += gridDim.x * blockDim.x) {
        const float4 u = p[i];
        const float4 v = p[(size_t)n4 + i];
        float4 r; r.x = u.x + v.x; r.y = u.y + v.y; r.z = u.z + v.z; r.w = u.w + v.w;
        o[i] = r;
    }
}

// ---------------------------------------------------------------------------
extern "C" void kernel_launch(void* const* d_in, const int* in_sizes, int n_in,
                              void* d_out, int out_size, void* d_ws, size_t ws_size,
                              hipStream_t stream) {
    const float* x  = (const float*)d_in[0];
    const float* gw = (const float*)d_in[1];
    const float* W0 = (const float*)d_in[2];
    const float* b0 = (const float*)d_in[3];
    const float* W1 = (const float*)d_in[4];
    const float* b1 = (const float*)d_in[5];
    const float* W2 = (const float*)d_in[6];
    const float* b2 = (const float*)d_in[7];
    float* out = (float*)d_out;

    const int T = in_sizes[0] / DDIM;   // 4096

    // workspace carve-up (256B aligned)
    char* wsb = (char*)d_ws;
    size_t off = 0;
    auto carve = [&](size_t bytes) -> void* {
        void* p = wsb + off;
        off = (off + bytes + 255) & ~(size_t)255;
        return p;
    };
    int*   counts   = (int*)  carve(E_EXPERTS * sizeof(int));
    int*   offsets  = (int*)  carve(E_EXPERTS * sizeof(int));
    int*   entries  = (int*)  carve((size_t)E_EXPERTS * T * sizeof(int));
    float* wlist    = (float*)carve((size_t)E_EXPERTS * T * sizeof(float));
    float* hbuf     = (float*)carve((size_t)TOPK * T * HDIM * sizeof(float)); // 64 MB
    float* pair_out = (float*)carve((size_t)TOPK * T * DDIM * sizeof(float)); // 32 MB
    (void)ws_size; (void)n_in; (void)out_size;

    zero_counts_kernel<<<1, 32, 0, stream>>>(counts);
    gate_kernel<<<(T + 3) / 4, 128, 0, stream>>>(x, gw, counts, entries, wlist, T);
    scan_kernel<<<1, 32, 0, stream>>>(counts, offsets);

    dim3 g1(HDIM / BN, (T + BM - 1) / BM, E_EXPERTS);   // 8 x 64 x 8
    ffn1_kernel<<<g1, 256, 0, stream>>>(x, W0, b0, W1, b1,
                                        counts, offsets, entries, hbuf, T);

    dim3 g2(DDIM / BN, (T + BM - 1) / BM, E_EXPERTS);   // 4 x 64 x 8
    ffn2_kernel<<<g2, 256, 0, stream>>>(hbuf, W2, b2, counts, offsets,
                                        entries, wlist, pair_out, T);

    const int n4 = (T * DDIM) / 4;
    combine_kernel<<<2048, 256, 0, stream>>>(pair_out, out, n4);
}